// AdaAT_18141941858815
// MI455X (gfx1250) — compile-verified
//
#include <hip/hip_runtime.h>

// ---------------------------------------------------------------------------
// Types for WMMA fragments (f32 16x16x4: A/B = 2 VGPRs, C/D = 8 VGPRs / lane)
// ---------------------------------------------------------------------------
typedef __attribute__((ext_vector_type(2))) float v2f;
typedef __attribute__((ext_vector_type(8))) float v8f;

#define LDS_STRIDE 257   // 257 mod 64 == 1 -> conflict-free column reads

__device__ __forceinline__ int clampi(int v, int lo, int hi) {
    return v < lo ? lo : (v > hi ? hi : v);
}

// A fragment (16x4, f32): lane L holds M = L&15; K = k0 + 2*(L>=16) + {0,1}
__device__ __forceinline__ v2f load_a_frag(const float* a_lds, int lane, int k0) {
    int m  = lane & 15;
    int kk = k0 + ((lane >> 4) << 1);
    v2f a;
    a.x = a_lds[m * LDS_STRIDE + kk];
    a.y = a_lds[m * LDS_STRIDE + kk + 1];
    return a;
}

// B fragment (4x16, f32): lane L holds N = n0 + (L&15); K = k0 + 2*(L>=16) + {0,1}
__device__ __forceinline__ v2f load_b_frag(const float* __restrict__ Wmat, int ncols,
                                           int k0, int n0, int lane) {
    int n  = n0 + (lane & 15);
    int kk = k0 + ((lane >> 4) << 1);
    v2f b;
    b.x = Wmat[kk * ncols + n];
    b.y = Wmat[(kk + 1) * ncols + n];
    return b;
}

// One 16x16 output tile over K=256 using chained f32 WMMA (exact f32 math).
__device__ __forceinline__ v8f wave_gemm_k256(const float* a_lds,
                                              const float* __restrict__ Wmat,
                                              int ncols, int n0, int lane) {
    v8f acc = {};
    #pragma unroll 4
    for (int k0 = 0; k0 < 256; k0 += 4) {
        v2f a = load_a_frag(a_lds, lane, k0);
        v2f b = load_b_frag(Wmat, ncols, k0, n0, lane);
        acc = __builtin_amdgcn_wmma_f32_16x16x4_f32(
            /*neg_a=*/false, a, /*neg_b=*/false, b,
            /*c_mod=*/(short)0, acc, /*reuse_a=*/false, /*reuse_b=*/false);
    }
    return acc;
}

// ---------------------------------------------------------------------------
// Kernel 1: MLP head.  1 block x 512 threads (16 waves).
//   Stage 1: p = relu(X @ W_c + b_c)         (M=16 padded, K=256, N=256)
//   Stage 2: p @ [W_s | W_r | W_t] + biases  (N = 256+256+512 = 64 tiles)
// Writes par_out[5][4][256]: scale, cos, sin, tx, ty  (b*256+d indexed).
// ---------------------------------------------------------------------------
__global__ __launch_bounds__(512) void adaat_head_kernel(
    const float* __restrict__ para,                  // (4,256)
    const float* __restrict__ Wc, const float* __restrict__ bc,
    const float* __restrict__ Ws, const float* __restrict__ bs,
    const float* __restrict__ Wr, const float* __restrict__ br,
    const float* __restrict__ Wt, const float* __restrict__ bt,
    float* __restrict__ par_out)
{
    __shared__ float a_lds[16 * LDS_STRIDE];
    __shared__ float p_lds[16 * LDS_STRIDE];

    const int tid  = threadIdx.x;
    const int lane = tid & 31;
    const int wave = tid >> 5;

    // Stage A into LDS: rows 0..3 = para_code, rows 4..15 = zero padding.
    for (int i = tid; i < 16 * 256; i += 512) {
        int m = i >> 8, k = i & 255;
        a_lds[m * LDS_STRIDE + k] = (m < 4) ? para[m * 256 + k] : 0.0f;
    }
    __syncthreads();

    // ---- Stage 1: wave w computes columns [16w, 16w+16) of p ----
    {
        const int n0 = wave * 16;
        v8f acc = wave_gemm_k256(a_lds, Wc, 256, n0, lane);
        const int n    = n0 + (lane & 15);
        const float bv = bc[n];
        const int mhi  = (lane >> 4) << 3;      // 0 or 8
        #pragma unroll
        for (int v = 0; v < 8; ++v) {
            float val = fmaxf(acc[v] + bv, 0.0f);   // ReLU
            p_lds[(mhi + v) * LDS_STRIDE + n] = val;
        }
    }
    __syncthreads();

    // ---- Stage 2: 64 N-tiles over [W_s | W_r | W_t] ----
    for (int t = wave; t < 64; t += 16) {
        const float* Wmat; const float* bias; int ncols; int nbase;
        if (t < 16)      { Wmat = Ws; bias = bs; ncols = 256; nbase = t * 16; }
        else if (t < 32) { Wmat = Wr; bias = br; ncols = 256; nbase = (t - 16) * 16; }
        else             { Wmat = Wt; bias = bt; ncols = 512; nbase = (t - 32) * 16; }

        v8f acc = wave_gemm_k256(p_lds, Wmat, ncols, nbase, lane);

        const int n    = nbase + (lane & 15);
        const float bv = bias[n];
        if (lane < 16) {                 // rows M = 0..7 live here; only M<4 real
            #pragma unroll
            for (int v = 0; v < 4; ++v) {
                float val = acc[v] + bv;
                if (t < 16) {
                    // scale = sigmoid(val) * 2
                    float sg = 1.0f / (1.0f + expf(-val));
                    par_out[v * 256 + n] = 2.0f * sg;
                } else if (t < 32) {
                    float ang = tanhf(val) * 3.14159f;
                    par_out[1024 + v * 256 + n] = cosf(ang);
                    par_out[2048 + v * 256 + n] = sinf(ang);
                } else {
                    // translation: col n = d*2 + comp
                    float tr = tanhf(val);
                    int dd = n >> 1, comp = n & 1;
                    par_out[(3 + comp) * 1024 + v * 256 + dd] = tr;
                }
            }
        }
    }
}

// ---------------------------------------------------------------------------
// Kernel 2: trilinear grid sample (align_corners=False semantics baked into
// the index math; zero padding via mask).  One thread per output element.
// Grid: (16, B*D) blocks x 256 threads; block => fixed (b,d) so the 5
// transform params are workgroup-uniform (scalar loads).
// ---------------------------------------------------------------------------
__global__ __launch_bounds__(256) void adaat_sample_kernel(
    const float* __restrict__ fm,     // (4,256,64,64)
    const float* __restrict__ par,    // 5 x 1024 params
    float* __restrict__ out)          // (4,256,64,64)
{
    const int bd = blockIdx.y;              // b*256 + d
    const int b  = bd >> 8;
    const int d  = bd & 255;
    const int idx = blockIdx.x * 256 + threadIdx.x;   // 0..4095 within slice
    const int h  = idx >> 6;
    const int w  = idx & 63;

    const float sc = par[bd];
    const float c  = par[1024 + bd];
    const float s  = par[2048 + bd];
    const float t0 = par[3072 + bd];
    const float t1 = par[4096 + bd];

    const float gx = 2.0f * (float)w * (1.0f / 63.0f)  - 1.0f;
    const float gy = 2.0f * (float)h * (1.0f / 63.0f)  - 1.0f;
    const float gz = 2.0f * (float)d * (1.0f / 255.0f) - 1.0f;

    const float tx = (c * gx - s * gy) * sc + t0;
    const float ty = (s * gx + c * gy) * sc + t1;

    const float ix = ((tx + 1.0f) * 64.0f  - 1.0f) * 0.5f;
    const float iy = ((ty + 1.0f) * 64.0f  - 1.0f) * 0.5f;
    const float iz = ((gz + 1.0f) * 256.0f - 1.0f) * 0.5f;

    const float x0f = floorf(ix), y0f = floorf(iy), z0f = floorf(iz);
    const float wx = ix - x0f, wy = iy - y0f, wz = iz - z0f;
    const int x0 = (int)x0f, y0 = (int)y0f, z0 = (int)z0f;

    const float* fmb = fm + (size_t)b * (256 * 64 * 64);

    auto G = [&](int z, int y, int x) -> float {
        bool ok = ((unsigned)x < 64u) && ((unsigned)y < 64u) && ((unsigned)z < 256u);
        int xc = clampi(x, 0, 63);
        int yc = clampi(y, 0, 63);
        int zc = clampi(z, 0, 255);
        float v = fmb[(zc << 12) + (yc << 6) + xc];
        return ok ? v : 0.0f;
    };

    const float uz = 1.0f - wz, uy = 1.0f - wy, ux = 1.0f - wx;
    float r =
        G(z0,     y0,     x0    ) * (uz * uy * ux) +
        G(z0,     y0,     x0 + 1) * (uz * uy * wx) +
        G(z0,     y0 + 1, x0    ) * (uz * wy * ux) +
        G(z0,     y0 + 1, x0 + 1) * (uz * wy * wx) +
        G(z0 + 1, y0,     x0    ) * (wz * uy * ux) +
        G(z0 + 1, y0,     x0 + 1) * (wz * uy * wx) +
        G(z0 + 1, y0 + 1, x0    ) * (wz * wy * ux) +
        G(z0 + 1, y0 + 1, x0 + 1) * (wz * wy * wx);

    out[(size_t)bd * 4096 + idx] = r;
}

// ---------------------------------------------------------------------------
// Launch
// ---------------------------------------------------------------------------
extern "C" void kernel_launch(void* const* d_in, const int* in_sizes, int n_in,
                              void* d_out, int out_size, void* d_ws, size_t ws_size,
                              hipStream_t stream) {
    const float* fm   = (const float*)d_in[0];   // feature_map (4,256,64,64)
    const float* para = (const float*)d_in[1];   // para_code   (4,256)
    const float* Wc   = (const float*)d_in[2];
    const float* bc   = (const float*)d_in[3];
    const float* Ws   = (const float*)d_in[4];
    const float* bs   = (const float*)d_in[5];
    const float* Wr   = (const float*)d_in[6];
    const float* br   = (const float*)d_in[7];
    const float* Wt   = (const float*)d_in[8];
    const float* bt   = (const float*)d_in[9];

    float* par = (float*)d_ws;                   // 5 * 1024 floats of scratch

    adaat_head_kernel<<<1, 512, 0, stream>>>(para, Wc, bc, Ws, bs, Wr, br, Wt, bt, par);

    dim3 grid(16, 4 * 256);                      // (HW/256, B*D)
    adaat_sample_kernel<<<grid, 256, 0, stream>>>(fm, par, (float*)d_out);
}